// EntityEmbedding_18433999634983
// MI455X (gfx1250) — compile-verified
//
#include <hip/hip_runtime.h>
#include <hip/hip_bf16.h>

// ---------------------------------------------------------------------------
// Reference: out[64] = relu( mean_{e: dst[e]==unseen} msg[e] ), where
//   msg[e, o] = sum_b att[edge_type[e], b] * ( x_cat[e, :] @ basis[b, :, o] )
//   x_cat[e]  = concat( entity_table[node_id[src[e]]], relation_embedding[rel_index[e]] )
// Only ~E/N (~20) edges have dst == unseen, so we filter + compact, then run a
// tiny dense WMMA (V_WMMA_F32_16X16X4_F32) over 16-edge tiles.
// ---------------------------------------------------------------------------

typedef __attribute__((ext_vector_type(2))) float v2f;
typedef __attribute__((ext_vector_type(8))) float v8f;

#define MAX_MATCH 4096
#define MAX_TILES (MAX_MATCH / 16)

// Workspace layout (bytes):
//   [0]      int   match_count
//   [64]     float acc_overflow[64]        (atomic fallback, normally untouched)
//   [320]    int   match_ids[MAX_MATCH]
//   [16704]  float part[MAX_TILES][64]     (deterministic per-tile partials)
#define WS_OFF_COUNT 0
#define WS_OFF_ACCOV 64
#define WS_OFF_IDS   320
#define WS_OFF_PART  (320 + MAX_MATCH * 4)
#define WS_BYTES     (WS_OFF_PART + MAX_TILES * 64 * 4)

// ---------------------------------------------------------------------------
__global__ void zero_ws_kernel(unsigned int* __restrict__ ws, int nwords) {
    for (int i = blockIdx.x * blockDim.x + threadIdx.x; i < nwords;
         i += gridDim.x * blockDim.x)
        ws[i] = 0u;
}

// ---------------------------------------------------------------------------
// Pass 1: scan all E edges (bandwidth bound: 4 MB of dst reads), compact
// matching edge ids. Overflow beyond MAX_MATCH handled by a scalar fallback
// so correctness holds even in pathological distributions.
__global__ void filter_kernel(const int* __restrict__ edge_index, int E,
                              const int* __restrict__ unseen_ptr,
                              int* __restrict__ count, int* __restrict__ ids,
                              const float* __restrict__ ent,
                              const float* __restrict__ relE,
                              const float* __restrict__ att,
                              const float* __restrict__ basis,
                              const int* __restrict__ node_id,
                              const int* __restrict__ edge_type,
                              const int* __restrict__ rel_index,
                              float* __restrict__ accOv) {
    const int unseen = *unseen_ptr;
    const int* __restrict__ dst = edge_index + E;
    for (long e = (long)blockIdx.x * blockDim.x + threadIdx.x; e < E;
         e += (long)gridDim.x * blockDim.x) {
        if (dst[e] != unseen) continue;
        int pos = atomicAdd(count, 1);
        if (pos < MAX_MATCH) {
            ids[pos] = (int)e;
        } else {
            // Scalar fallback: compute this edge's message directly.
            int s = edge_index[e];
            const float* xrow = ent + (long)node_id[s] * 64;
            const float* rrow = relE + (long)rel_index[e] * 32;
            float c0 = att[(long)edge_type[e] * 2 + 0];
            float c1 = att[(long)edge_type[e] * 2 + 1];
            for (int o = 0; o < 64; ++o) {
                float s0 = 0.f, s1 = 0.f;
                for (int i = 0; i < 64; ++i) {
                    float xv = xrow[i];
                    s0 += xv * basis[(0 * 96 + i) * 64 + o];
                    s1 += xv * basis[(1 * 96 + i) * 64 + o];
                }
                for (int i = 0; i < 32; ++i) {
                    float rv = rrow[i];
                    s0 += rv * basis[(0 * 96 + 64 + i) * 64 + o];
                    s1 += rv * basis[(1 * 96 + 64 + i) * 64 + o];
                }
                atomicAdd(&accOv[o], c0 * s0 + c1 * s1);
            }
        }
    }
}

// ---------------------------------------------------------------------------
// Pass 2: one wave32 per 16-edge tile. Builds the 16x96 x_cat tile in LDS,
// then for each basis b and each 16-wide output tile computes
//   Y = A(16x96) @ basis[b](96x64-slice)  via 24 chained V_WMMA_F32_16X16X4_F32,
// applies att coefficients per edge, reduces over the 16 edges, and writes a
// deterministic per-tile partial sum.
//
// F32 16x16x4 fragment layout (ISA 7.12.2):
//   A: lane L(<16) holds A[M=L,K=k0..k0+1] in v0/v1; lane L+16 holds K=k0+2..3
//   B: mirrored over K rows (v0: K=k0 / K=k0+2 ; v1: K=k0+1 / K=k0+3)
//   D: lane L holds N=L%16; vgpr j holds M=j (lanes<16) / M=j+8 (lanes>=16)
__global__ __launch_bounds__(32) void wmma_msg_kernel(
    const float* __restrict__ ent, const float* __restrict__ relE,
    const float* __restrict__ att, const float* __restrict__ basis,
    const int* __restrict__ node_id, const int* __restrict__ edge_index,
    const int* __restrict__ edge_type, const int* __restrict__ rel_index,
    int E, const int* __restrict__ count, const int* __restrict__ ids,
    float* __restrict__ part) {
    __shared__ float Xt[16 * 97];   // padded stride 97 to dodge bank conflicts
    __shared__ float coefS[16 * 2];

    const int lane = threadIdx.x;          // 0..31
    int mc = *count;
    if (mc > MAX_MATCH) mc = MAX_MATCH;
    const int tiles = (mc + 15) >> 4;

    for (int t = blockIdx.x; t < tiles; t += gridDim.x) {
        const int valid = min(16, mc - t * 16);

        // Stage x_cat tile [16 x 96] into LDS (zeros for invalid rows).
        for (int idx = lane; idx < 16 * 96; idx += 32) {
            int m = idx / 96;
            int c = idx - m * 96;
            float v = 0.f;
            if (m < valid) {
                int e = ids[t * 16 + m];
                if (c < 64) {
                    int s = edge_index[e];  // src
                    v = ent[(long)node_id[s] * 64 + c];
                } else {
                    v = relE[(long)rel_index[e] * 32 + (c - 64)];
                }
            }
            Xt[m * 97 + c] = v;
        }
        // Stage coefficients att[edge_type[e], b].
        {
            int m = lane >> 1, bb = lane & 1;
            float cv = 0.f;
            if (m < valid) {
                int e = ids[t * 16 + m];
                cv = att[(long)edge_type[e] * 2 + bb];
            }
            coefS[m * 2 + bb] = cv;
        }
        __syncthreads();

        const int half = lane >> 4;     // 0 or 1
        const int mlo  = lane & 15;

        for (int nt = 0; nt < 4; ++nt) {
            float colsum = 0.f;
            for (int bb = 0; bb < 2; ++bb) {
                v8f c = {};
#pragma unroll
                for (int k0 = 0; k0 < 96; k0 += 4) {
                    v2f a, bf;
                    a.x = Xt[mlo * 97 + k0 + 2 * half];
                    a.y = Xt[mlo * 97 + k0 + 2 * half + 1];
                    const float* bp =
                        basis + ((long)bb * 96 + k0 + 2 * half) * 64 + nt * 16 + mlo;
                    bf.x = bp[0];
                    bf.y = bp[64];
                    c = __builtin_amdgcn_wmma_f32_16x16x4_f32(
                        false, a, false, bf, (short)0, c, false, false);
                }
                // Weight by this edge's att coefficient and reduce over M.
#pragma unroll
                for (int j = 0; j < 8; ++j)
                    colsum += coefS[(j + 8 * half) * 2 + bb] * c[j];
            }
            // Combine the two lane-halves (M 0..7 with M 8..15, same N).
            colsum += __shfl_xor(colsum, 16, 32);
            if (lane < 16)
                part[(long)t * 64 + nt * 16 + lane] = colsum;
        }
        __syncthreads();
    }
}

// ---------------------------------------------------------------------------
__global__ void finalize_kernel(const float* __restrict__ part,
                                const float* __restrict__ accOv,
                                const int* __restrict__ count,
                                float* __restrict__ out) {
    int o = threadIdx.x;  // 64 threads
    int mc = *count;
    int tiles = (min(mc, MAX_MATCH) + 15) >> 4;
    float s = accOv[o];
    for (int t = 0; t < tiles; ++t) s += part[t * 64 + o];
    float cnt = (mc > 0) ? (float)mc : 1.0f;  // jnp.maximum(cnt, 1.0)
    float v = s / cnt;
    out[o] = v > 0.f ? v : 0.f;               // ReLU
}

// ---------------------------------------------------------------------------
extern "C" void kernel_launch(void* const* d_in, const int* in_sizes, int n_in,
                              void* d_out, int out_size, void* d_ws,
                              size_t ws_size, hipStream_t stream) {
    const float* ent       = (const float*)d_in[0];  // [N_ENT, 64]
    const float* relE      = (const float*)d_in[1];  // [R, 32]
    const float* att       = (const float*)d_in[2];  // [2R, 2]
    const float* basis     = (const float*)d_in[3];  // [2, 96, 64]
    const int*   node_id   = (const int*)d_in[4];    // [N]
    const int*   edge_idx  = (const int*)d_in[5];    // [2, E] flat
    const int*   edge_type = (const int*)d_in[6];    // [E]
    const int*   rel_index = (const int*)d_in[7];    // [E]
    const int*   unseen    = (const int*)d_in[8];    // scalar

    const int E = in_sizes[6];

    char* ws = (char*)d_ws;
    int*   ws_count = (int*)(ws + WS_OFF_COUNT);
    float* ws_accov = (float*)(ws + WS_OFF_ACCOV);
    int*   ws_ids   = (int*)(ws + WS_OFF_IDS);
    float* ws_part  = (float*)(ws + WS_OFF_PART);

    // 1) Zero workspace (count, overflow acc, partials).
    {
        int nwords = WS_BYTES / 4;
        int blocks = (nwords + 255) / 256;
        zero_ws_kernel<<<blocks, 256, 0, stream>>>((unsigned int*)d_ws, nwords);
    }
    // 2) Filter + compact edges with dst == unseen (bandwidth bound, 4 MB).
    filter_kernel<<<2048, 256, 0, stream>>>(edge_idx, E, unseen, ws_count,
                                            ws_ids, ent, relE, att, basis,
                                            node_id, edge_type, rel_index,
                                            ws_accov);
    // 3) WMMA over compacted 16-edge tiles (expected ~2 tiles of work).
    wmma_msg_kernel<<<64, 32, 0, stream>>>(ent, relE, att, basis, node_id,
                                           edge_idx, edge_type, rel_index, E,
                                           ws_count, ws_ids, ws_part);
    // 4) Mean + ReLU for the single requested row.
    finalize_kernel<<<1, 64, 0, stream>>>(ws_part, ws_accov, ws_count,
                                          (float*)d_out);
}